// IterativeQuadInterp3d_63453846831184
// MI455X (gfx1250) — compile-verified
//
#include <hip/hip_runtime.h>
#include <cstdint>
#include <cstddef>

// ---------------------------------------------------------------------------
// IterativeQuadInterp3d for MI455X (gfx1250, wave32).
// Memory-bound stencil: 80 MB of mandatory HBM traffic -> ~3.4 us floor at
// 23.3 TB/s. Strategy: async global->LDS staging of haloed tiles (ASYNCcnt
// path) for the dense 27-tap NMS; sparse (~3.7%) iterative refinement reads
// from global (input is fully L2-resident: 16 MB vs 192 MB L2).
// ---------------------------------------------------------------------------

namespace {
constexpr int kB  = 2;
constexpr int kC  = 4;
constexpr int kBC = kB * kC;          // 8
constexpr int kD  = 8;
constexpr int kH  = 256;
constexpr int kW  = 256;
constexpr int kHW  = kH * kW;         // 65536
constexpr int kDHW = kD * kHW;        // 524288

constexpr int   TILE  = 16;           // H x W tile per block
constexpr int   LTH   = TILE + 2;     // 18 (halo 1)
constexpr int   LTD   = kD + 2;       // 10 (halo 1 in depth, edge-clamped)
constexpr int   LDS_ELEMS = LTD * LTH * LTH;   // 3240 floats = 12.96 KB
constexpr int   NITERS = 5;
constexpr float MAXSH  = 0.6f;
constexpr float BONUS  = 10.0f;
}

__device__ __forceinline__ int iclamp(int v, int lo, int hi) {
    return v < lo ? lo : (v > hi ? hi : v);
}

// One float, global -> LDS, asynchronous (tracked by ASYNCcnt).
__device__ __forceinline__ void async_copy_f32(const float* gsrc, float* ldst) {
#if defined(__has_builtin) && __has_builtin(__builtin_amdgcn_global_load_async_to_lds_b32)
    typedef __attribute__((address_space(1))) int* as1_ip;
    typedef __attribute__((address_space(3))) int* as3_ip;
    __builtin_amdgcn_global_load_async_to_lds_b32(
        (as1_ip)(void*)gsrc, (as3_ip)(void*)ldst, 0, 0);
#else
    // Flat LDS pointer's low 32 bits are the LDS byte offset (aperture form).
    unsigned lds_off = (unsigned)(size_t)(void*)ldst;
    asm volatile("global_load_async_to_lds_b32 %0, %1, off"
                 :: "v"(lds_off), "v"(gsrc)
                 : "memory");
#endif
}

__device__ __forceinline__ void wait_async_zero() {
#if defined(__has_builtin) && __has_builtin(__builtin_amdgcn_s_wait_asynccnt)
    __builtin_amdgcn_s_wait_asynccnt(0);
#else
    asm volatile("s_wait_asynccnt 0x0" ::: "memory");
#endif
}

__global__ __launch_bounds__(256)
void IterativeQuadInterp3d_kernel(const float* __restrict__ x,
                                  float* __restrict__ coords,   // (BC,3,D,H,W)
                                  float* __restrict__ ymax) {   // (BC,D,H,W)
    __shared__ float tile[LTD][LTH][LTH];

    const int tid    = threadIdx.x;
    const int w_base = blockIdx.x * TILE;
    const int h_base = blockIdx.y * TILE;
    const int bc     = blockIdx.z;

    const float* vol = x + (size_t)bc * kDHW;

    // -------- stage edge-clamped haloed tile into LDS (async pipe) --------
    float* lbase = &tile[0][0][0];
    for (int e = tid; e < LDS_ELEMS; e += 256) {
        int zd = e / (LTH * LTH);
        int r  = e % (LTH * LTH);
        int zh = r / LTH;
        int zw = r % LTH;
        int gd = iclamp(zd - 1, 0, kD - 1);                 // edge padding
        int gh = iclamp(h_base + zh - 1, 0, kH - 1);
        int gw = iclamp(w_base + zw - 1, 0, kW - 1);
        async_copy_f32(vol + (size_t)gd * kHW + (size_t)gh * kW + gw, lbase + e);
    }
    wait_async_zero();     // this wave's async loads landed in LDS
    __syncthreads();       // all waves' loads landed

    const int lx = tid & (TILE - 1);
    const int ly = tid >> 4;
    const int w0 = w_base + lx;
    const int h0 = h_base + ly;

    for (int d0 = 0; d0 < kD; ++d0) {
        // ---------------- 26-neighbor NMS from LDS ----------------
        float c = tile[d0 + 1][ly + 1][lx + 1];
        float nmax = -__FLT_MAX__;
#pragma unroll
        for (int dd = 0; dd < 3; ++dd)
#pragma unroll
            for (int dh = 0; dh < 3; ++dh)
#pragma unroll
                for (int dw = 0; dw < 3; ++dw) {
                    if (dd == 1 && dh == 1 && dw == 1) continue;
                    nmax = fmaxf(nmax, tile[d0 + dd][ly + dh][lx + dw]);
                }
        const bool mask = c > nmax;

        const size_t vox   = (size_t)d0 * kHW + (size_t)h0 * kW + w0;
        const size_t cbase = (size_t)bc * 3 * kDHW + vox;

        float out_s = (float)d0;   // channel 0: depth coord
        float out_x = (float)w0;   // channel 1: width coord
        float out_y = (float)h0;   // channel 2: height coord
        float yv    = c;

        if (mask) {
            // -------- 5-iteration quadratic refinement (sparse path) --------
            int  d_cur = d0, h_cur = h0, w_cur = w0;
            bool valid = true;
            float shx = 0.f, shy = 0.f, shs = 0.f, gds = 0.f;

            for (int it = 0; it < NITERS; ++it) {
                const int d_s = iclamp(d_cur, 1, kD - 2);
                const int h_s = iclamp(h_cur, 1, kH - 2);
                const int w_s = iclamp(w_cur, 1, kW - 2);
                const float* p = vol + (size_t)d_s * kHW + (size_t)h_s * kW + w_s;

                const float c000 = p[0];
                const float pxm = p[-1],    pxp = p[1];
                const float pym = p[-kW],   pyp = p[kW];
                const float psm = p[-kHW],  psp = p[kHW];

                const float gx = 0.5f * (pxp - pxm);
                const float gy = 0.5f * (pyp - pym);
                const float gs = 0.5f * (psp - psm);
                const float dxx = pxp - 2.f * c000 + pxm;
                const float dyy = pyp - 2.f * c000 + pym;
                const float dss = psp - 2.f * c000 + psm;
                const float dxy = 0.25f * (p[ kW + 1] - p[ kW - 1] - p[-kW + 1] + p[-kW - 1]);
                const float dxs = 0.25f * (p[ kHW + 1] - p[ kHW - 1] - p[-kHW + 1] + p[-kHW - 1]);
                const float dys = 0.25f * (p[ kHW + kW] - p[ kHW - kW] - p[-kHW + kW] + p[-kHW - kW]);

                const float cf00 = dyy * dss - dys * dys;
                const float cf01 = dxy * dss - dys * dxs;
                const float cf02 = dxy * dys - dyy * dxs;
                const float det  = dxx * cf00 - dxy * cf01 + dxs * cf02;
                const bool  solved = fabsf(det) > 0.f;
                const float safe = solved ? det : 1.f;
                const float r0 = -gx, r1 = -gy, r2 = -gs;
                float sx = (r0 * cf00 - dxy * (r1 * dss - dys * r2) + dxs * (r1 * dys - dyy * r2)) / safe;
                float sy = (dxx * (r1 * dss - dys * r2) - r0 * cf01 + dxs * (dxy * r2 - r1 * dxs)) / safe;
                float ss = (dxx * (dyy * r2 - r1 * dys) - dxy * (dxy * r2 - r1 * dxs) + r0 * cf02) / safe;

                valid = valid && solved;
                const float vf = valid ? 1.f : 0.f;
                sx *= vf; sy *= vf; ss *= vf;
                if (valid) {
                    shx = sx; shy = sy; shs = ss;
                    gds = gx * sx + gy * sy + gs * ss;
                }

                int mv = (valid && sx >  MAXSH) ? 1 : ((valid && sx < -MAXSH) ? -1 : 0);
                int nw = w_cur + mv;
                valid = valid && nw >= 1 && nw <= kW - 2;
                w_cur = iclamp(nw, 0, kW - 1);

                mv = (valid && sy >  MAXSH) ? 1 : ((valid && sy < -MAXSH) ? -1 : 0);
                int nh = h_cur + mv;
                valid = valid && nh >= 1 && nh <= kH - 2;
                h_cur = iclamp(nh, 0, kH - 1);

                mv = (valid && ss >  MAXSH) ? 1 : ((valid && ss < -MAXSH) ? -1 : 0);
                int nd = d_cur + mv;
                valid = valid && nd >= 1 && nd <= kD - 2;
                d_cur = iclamp(nd, 0, kD - 1);
            }

            valid = valid && fabsf(shx) <= 1.5f && fabsf(shy) <= 1.5f && fabsf(shs) <= 1.5f;
            if (valid) {
                out_s = (float)d_cur + shs;
                out_x = (float)w_cur + shx;
                out_y = (float)h_cur + shy;
                yv    = c + 0.5f * gds + BONUS;
            }
            // mask && !valid: coords fall back to base, corr == 0 -> yv = c.
        }

        coords[cbase]            = out_s;
        coords[cbase +     kDHW] = out_x;
        coords[cbase + 2 * kDHW] = out_y;
        ymax[(size_t)bc * kDHW + vox] = yv;
    }
}

extern "C" void kernel_launch(void* const* d_in, const int* in_sizes, int n_in,
                              void* d_out, int out_size, void* d_ws, size_t ws_size,
                              hipStream_t stream) {
    (void)in_sizes; (void)n_in; (void)out_size; (void)d_ws; (void)ws_size;
    const float* x = (const float*)d_in[0];
    float* coords = (float*)d_out;                          // (B,C,3,D,H,W)
    float* ymax   = coords + (size_t)kBC * 3 * kDHW;        // (B,C,D,H,W)

    dim3 grid(kW / TILE, kH / TILE, kBC);                   // (16,16,8)
    dim3 block(256);                                        // 8 x wave32
    IterativeQuadInterp3d_kernel<<<grid, block, 0, stream>>>(x, coords, ymax);
}